// TopKRouter_22316650070633
// MI455X (gfx1250) — compile-verified
//
#include <hip/hip_runtime.h>

typedef __attribute__((ext_vector_type(2))) float v2f;
typedef __attribute__((ext_vector_type(4))) float v4f;
typedef __attribute__((ext_vector_type(8))) float v8f;
typedef int v4i_b128 __attribute__((vector_size(16)));   // matches builtin param type

#define M_TOT   16384       // B*S tokens
#define DDIM    2048        // hidden dim (K of GEMM)
#define EXPERTS 64          // experts (N of GEMM)
#define KSEL    8           // top-k
#define MT      32          // tokens per block (2 M-subtiles per wave)
#define KC      256         // K chunk staged in LDS (double buffered)
#define XS_STRIDE (KC + 4)  // bank = (4*row + k) % 64 -> conflict-free A reads
#define SC_STRIDE (EXPERTS + 4)

#if defined(__gfx1250__) && __has_builtin(__builtin_amdgcn_global_load_async_to_lds_b128)
#define USE_ASYNC_LDS 1
#else
#define USE_ASYNC_LDS 0
#endif

__device__ __forceinline__ void wait_async_zero() {
#if USE_ASYNC_LDS
#if __has_builtin(__builtin_amdgcn_s_wait_asynccnt)
    __builtin_amdgcn_s_wait_asynccnt(0);
#else
    asm volatile("s_wait_asynccnt 0x0" ::: "memory");
#endif
#endif
}

__global__ __launch_bounds__(128)
void TopKRouter_fused_kernel(const float* __restrict__ x,
                             const float* __restrict__ W,
                             const float* __restrict__ bias,
                             float* __restrict__ out) {
    __shared__ float xs[2][MT * XS_STRIDE];            // double-buffered x chunks (~66 KB)
    __shared__ float sc[MT * SC_STRIDE];               // 32x64 scores
    __shared__ unsigned long long sel_s[MT];
    __shared__ float mx_s[MT];
    __shared__ float inv_s[MT];

    const int tid  = threadIdx.x;
    const int lane = tid & 31;
    const int wave = tid >> 5;          // 0..3 -> expert tile
    const int m0   = blockIdx.x * MT;   // token base
    const int n0   = wave * 16;         // expert base

    const int lrow = lane & 15;         // 0..15
    const int lhi  = lane >> 4;         // 0/1 -> K offset +2 for hi half

    v8f acc0 = {};                      // tokens m0..m0+15
    v8f acc1 = {};                      // tokens m0+16..m0+31

    // Stage one [MT x KC] x-chunk into xs[buf]: 2048 x 16B transfers / 128 threads.
    // Async path: HBM -> LDS directly (no VGPR round-trip), tracked by ASYNCcnt.
    auto stage = [&](int kc, int buf) {
        #pragma unroll 4
        for (int i = 0; i < MT * (KC / 4) / 128; ++i) {
            const int idx = tid + i * 128;
            const int row = idx >> 6;        // / (KC/4)
            const int c4  = idx & 63;        // % (KC/4)
            const float* gp = x + (size_t)(m0 + row) * DDIM + kc + c4 * 4;
            float* lp = &xs[buf][row * XS_STRIDE + c4 * 4];
#if USE_ASYNC_LDS
            __builtin_amdgcn_global_load_async_to_lds_b128(
                (__attribute__((address_space(1))) v4i_b128*)gp,
                (__attribute__((address_space(3))) v4i_b128*)lp,
                /*offset=*/0, /*cpol=*/0);
#else
            v4f v = __builtin_nontemporal_load((const v4f*)gp);
            *(v4f*)lp = v;
#endif
        }
    };

    stage(0, 0);
    int buf = 0;

    for (int kc = 0; kc < DDIM; kc += KC) {
        wait_async_zero();     // this wave's async fills of xs[buf] done
        __syncthreads();       // everyone's fills done; also fences reuse of xs[buf]

        if (kc + KC < DDIM) stage(kc + KC, buf ^ 1);   // overlap next chunk with compute

        const float* xb = &xs[buf][0];
        #pragma unroll 4
        for (int k = 0; k < KC; k += 4) {
            const int kk = k + 2 * lhi;
            // A fragments for the two M-subtiles (16x4 f32 each)
            v2f a0 = *(const v2f*)(&xb[lrow * XS_STRIDE + kk]);
            v2f a1 = *(const v2f*)(&xb[(16 + lrow) * XS_STRIDE + kk]);
            // One B fragment (4x16 of W, L2-resident) shared by both WMMAs
            const size_t krow = (size_t)(kc + kk);
            v2f bf;
            bf.x = W[krow * EXPERTS + n0 + lrow];
            bf.y = W[(krow + 1) * EXPERTS + n0 + lrow];
            acc0 = __builtin_amdgcn_wmma_f32_16x16x4_f32(
                false, a0, false, bf, (short)0, acc0, false, false);
            acc1 = __builtin_amdgcn_wmma_f32_16x16x4_f32(
                false, a1, false, bf, (short)0, acc1, false, false);
        }
        buf ^= 1;
    }

    // bias + ReLU, stage scores in LDS
    // C layout: vgpr r, lane l -> m = r + 8*(l>>4), n = l&15
    const float bv = bias[n0 + lrow];
    #pragma unroll
    for (int r = 0; r < 8; ++r) {
        const int m = r + 8 * lhi;
        float s0 = acc0[r] + bv; s0 = s0 > 0.0f ? s0 : 0.0f;
        float s1 = acc1[r] + bv; s1 = s1 > 0.0f ? s1 : 0.0f;
        sc[m * SC_STRIDE + n0 + lrow]        = s0;
        sc[(16 + m) * SC_STRIDE + n0 + lrow] = s1;
    }
    __syncthreads();

    // top-8 selection: one lane per token (strict '>' = lax.top_k tie-break)
    if (tid < MT) {
        const float* row = &sc[tid * SC_STRIDE];
        unsigned long long sel = 0ull;
        float vals[KSEL];
        for (int i = 0; i < KSEL; ++i) {
            float best = -1.0f;   // scores >= 0 after ReLU
            int   bi   = 0;
            for (int e = 0; e < EXPERTS; ++e) {
                const float v = row[e];
                if (!((sel >> e) & 1ull) && v > best) { best = v; bi = e; }
            }
            sel |= 1ull << bi;
            vals[i] = best;
        }
        const float mx = vals[0];
        float sum = 0.0f;
        #pragma unroll
        for (int i = 0; i < KSEL; ++i) sum += __expf(vals[i] - mx);
        sel_s[tid] = sel;
        mx_s[tid]  = mx;
        inv_s[tid] = 1.0f / sum;
    }
    __syncthreads();

    // cooperative coalesced epilogue: 4 threads x 16 consecutive floats per token
    {
        const int t  = tid >> 2;         // token 0..31
        const int e0 = (tid & 3) * 16;   // expert base within row
        const unsigned long long sel = sel_s[t];
        const float mx  = mx_s[t];
        const float inv = inv_s[t];
        const float* row = &sc[t * SC_STRIDE];
        const size_t gbase = (size_t)(m0 + t) * EXPERTS + e0;
        float* __restrict__ rw = out + gbase;                           // router_weight
        float* __restrict__ mk = out + (size_t)M_TOT * EXPERTS + gbase; // mask
        #pragma unroll 4
        for (int j = 0; j < 16; ++j) {
            const int e = e0 + j;
            const bool s = (sel >> e) & 1ull;
            __builtin_nontemporal_store(s ? 1.0f : 0.0f, mk + j);
            __builtin_nontemporal_store(s ? __expf(row[e] - mx) * inv : 0.0f, rw + j);
        }
    }
}

extern "C" void kernel_launch(void* const* d_in, const int* in_sizes, int n_in,
                              void* d_out, int out_size, void* d_ws, size_t ws_size,
                              hipStream_t stream) {
    const float* x = (const float*)d_in[0];
    const float* W = (const float*)d_in[1];
    const float* b = (const float*)d_in[2];
    float* out     = (float*)d_out;

    dim3 grid(M_TOT / MT);
    dim3 block(128);
    TopKRouter_fused_kernel<<<grid, block, 0, stream>>>(x, W, b, out);
}